// PillarFeatureNet_34729105555637
// MI455X (gfx1250) — compile-verified
//
#include <hip/hip_runtime.h>

typedef float v2f __attribute__((ext_vector_type(2)));
typedef float v8f __attribute__((ext_vector_type(8)));

#define P_PILLARS 60000
#define NPT 32
#define COUT 64
#define WPB 8                      // waves (pillars) per block
#define NBLK_A (P_PILLARS / WPB)   // 7500

// Pass A: build 9-dim feats, GEMM via WMMA f32 16x16x4, per-pillar max/min,
// per-block partial sum / sumsq for batch-norm statistics.
__global__ __launch_bounds__(256) void pfn_pass_a(
    const float* __restrict__ features,   // [P,32,4]
    const int*   __restrict__ num_points, // [P]
    const int*   __restrict__ coors,      // [P,4] (b,z,y,x)
    const float* __restrict__ W,          // [9,64]
    float* __restrict__ ws_mx, float* __restrict__ ws_mn,   // [P,64] each
    float* __restrict__ ws_psum, float* __restrict__ ws_pssq) // [NBLK_A,64]
{
    __shared__ float Wl[12][COUT];              // K padded 9 -> 12
    __shared__ float featsL[WPB][NPT][12];
    __shared__ float wsum[WPB][COUT];
    __shared__ float wssq[WPB][COUT];

    const int tid = threadIdx.x;
    for (int i = tid; i < 12 * COUT; i += 256) {
        const int r = i / COUT, c = i % COUT;
        Wl[r][c] = (r < 9) ? W[r * COUT + c] : 0.0f;
    }

    const int wid  = tid >> 5;
    const int lane = tid & 31;
    const int half = lane >> 4;   // lane group (0: lanes 0-15, 1: lanes 16-31)
    const int l16  = lane & 15;
    const int p    = blockIdx.x * WPB + wid;   // P % WPB == 0 -> always valid

    // ---- per-point raw features (lane = point index) ----
    const float4 f = ((const float4*)features)[p * NPT + lane];
    const int npts = num_points[p];

    // reference sums ALL 32 points (no mask) then divides by num_points
    float sx = f.x, sy = f.y, sz = f.z;
    #pragma unroll
    for (int off = 16; off >= 1; off >>= 1) {
        sx += __shfl_xor(sx, off, 32);
        sy += __shfl_xor(sy, off, 32);
        sz += __shfl_xor(sz, off, 32);
    }
    const float inv_n = 1.0f / (float)npts;
    const float cmx = sx * inv_n, cmy = sy * inv_n, cmz = sz * inv_n;

    const float xc = (float)coors[p * 4 + 3] * 0.2f + 0.1f;           // vx/2 + 0
    const float yc = (float)coors[p * 4 + 2] * 0.2f + (0.1f - 40.0f); // vy/2 - 40

    const float m = (lane < npts) ? 1.0f : 0.0f;   // zero padded rows AFTER offsets
    float* row = &featsL[wid][lane][0];
    row[0] = f.x * m;           row[1] = f.y * m;
    row[2] = f.z * m;           row[3] = f.w * m;
    row[4] = (f.x - cmx) * m;   row[5] = (f.y - cmy) * m;
    row[6] = (f.z - cmz) * m;   row[7] = (f.x - xc) * m;
    row[8] = (f.y - yc) * m;
    row[9] = 0.0f; row[10] = 0.0f; row[11] = 0.0f;

    __syncthreads();

    // ---- B fragments (ISA 16x4 layout mirrored: K rows striped by lane half) ----
    v2f bfr[3][4];
    #pragma unroll
    for (int ks = 0; ks < 3; ++ks) {
        const int kr = ks * 4 + 2 * half;
        #pragma unroll
        for (int nt = 0; nt < 4; ++nt) {
            bfr[ks][nt][0] = Wl[kr][nt * 16 + l16];
            bfr[ks][nt][1] = Wl[kr + 1][nt * 16 + l16];
        }
    }

    float chmax[4], chmin[4], chs[4], chss[4];
    #pragma unroll
    for (int nt = 0; nt < 4; ++nt) {
        chmax[nt] = -3.402823e38f; chmin[nt] = 3.402823e38f;
        chs[nt] = 0.0f; chss[nt] = 0.0f;
    }

    #pragma unroll
    for (int mt = 0; mt < 2; ++mt) {          // 32 points = 2 M-tiles
        v8f acc[4];
        #pragma unroll
        for (int nt = 0; nt < 4; ++nt) acc[nt] = (v8f){0,0,0,0,0,0,0,0};
        #pragma unroll
        for (int ks = 0; ks < 3; ++ks) {      // K = 12 (padded) = 3 steps of 4
            v2f a;
            a[0] = featsL[wid][mt * 16 + l16][ks * 4 + 2 * half];
            a[1] = featsL[wid][mt * 16 + l16][ks * 4 + 2 * half + 1];
            #pragma unroll
            for (int nt = 0; nt < 4; ++nt) {  // 64 channels = 4 N-tiles
                acc[nt] = __builtin_amdgcn_wmma_f32_16x16x4_f32(
                    false, a, false, bfr[ks][nt], (short)0, acc[nt], false, false);
            }
        }
        #pragma unroll
        for (int nt = 0; nt < 4; ++nt) {
            #pragma unroll
            for (int r = 0; r < 8; ++r) {
                const float v = acc[nt][r];
                chmax[nt] = fmaxf(chmax[nt], v);
                chmin[nt] = fminf(chmin[nt], v);
                chs[nt]  += v;
                chss[nt] += v * v;
            }
        }
    }
    // combine lane halves (rows r and r+8 live in opposite halves)
    #pragma unroll
    for (int nt = 0; nt < 4; ++nt) {
        chmax[nt] = fmaxf(chmax[nt], __shfl_xor(chmax[nt], 16, 32));
        chmin[nt] = fminf(chmin[nt], __shfl_xor(chmin[nt], 16, 32));
        chs[nt]  += __shfl_xor(chs[nt], 16, 32);
        chss[nt] += __shfl_xor(chss[nt], 16, 32);
    }
    if (half == 0) {
        #pragma unroll
        for (int nt = 0; nt < 4; ++nt) {
            const int c = nt * 16 + l16;
            ws_mx[(size_t)p * COUT + c] = chmax[nt];
            ws_mn[(size_t)p * COUT + c] = chmin[nt];
            wsum[wid][c] = chs[nt];
            wssq[wid][c] = chss[nt];
        }
    }
    __syncthreads();
    if (tid < COUT) {   // fixed-order block reduction -> deterministic
        float s = 0.0f, ss = 0.0f;
        #pragma unroll
        for (int w = 0; w < WPB; ++w) { s += wsum[w][tid]; ss += wssq[w][tid]; }
        ws_psum[(size_t)blockIdx.x * COUT + tid] = s;
        ws_pssq[(size_t)blockIdx.x * COUT + tid] = ss;
    }
}

// Pass B: reduce per-block partials (fixed order), produce BN scale/shift.
__global__ void pfn_stats(const float* __restrict__ psum,
                          const float* __restrict__ pssq,
                          const float* __restrict__ gamma,
                          const float* __restrict__ beta,
                          float* __restrict__ scale,
                          float* __restrict__ shift)
{
    const int c = threadIdx.x;
    if (c >= COUT) return;
    float s = 0.0f, ss = 0.0f;
    for (int b = 0; b < NBLK_A; ++b) {
        s  += psum[(size_t)b * COUT + c];
        ss += pssq[(size_t)b * COUT + c];
    }
    const float inv = 1.0f / (float)(P_PILLARS * NPT);
    const float mean = s * inv;
    const float var  = ss * inv - mean * mean;   // biased, matches torch BN train
    const float sc   = gamma[c] * rsqrtf(var + 1e-3f);
    scale[c] = sc;
    shift[c] = beta[c] - mean * sc;
}

// Pass C: out = relu(scale * (scale>=0 ? max : min) + shift)
__global__ __launch_bounds__(256) void pfn_finish(
    const float* __restrict__ ws_mx, const float* __restrict__ ws_mn,
    const float* __restrict__ scale, const float* __restrict__ shift,
    float* __restrict__ out)
{
    const int idx = blockIdx.x * 256 + threadIdx.x;  // P*64 total
    const int c = idx & (COUT - 1);
    const float s = scale[c];
    const float v = (s >= 0.0f) ? ws_mx[idx] : ws_mn[idx];
    out[idx] = fmaxf(fmaf(s, v, shift[c]), 0.0f);
}

extern "C" void kernel_launch(void* const* d_in, const int* in_sizes, int n_in,
                              void* d_out, int out_size, void* d_ws, size_t ws_size,
                              hipStream_t stream) {
    const float* features   = (const float*)d_in[0];
    const int*   num_points = (const int*)  d_in[1];
    const int*   coors      = (const int*)  d_in[2];
    const float* W          = (const float*)d_in[3];
    const float* gamma      = (const float*)d_in[4];
    const float* beta       = (const float*)d_in[5];
    float* out = (float*)d_out;

    float* ws      = (float*)d_ws;
    float* ws_mx   = ws;                                    // P*64
    float* ws_mn   = ws_mx   + (size_t)P_PILLARS * COUT;    // P*64
    float* ws_psum = ws_mn   + (size_t)P_PILLARS * COUT;    // NBLK_A*64
    float* ws_pssq = ws_psum + (size_t)NBLK_A * COUT;       // NBLK_A*64
    float* ws_scl  = ws_pssq + (size_t)NBLK_A * COUT;       // 64
    float* ws_shf  = ws_scl + COUT;                         // 64

    pfn_pass_a<<<NBLK_A, 256, 0, stream>>>(features, num_points, coors, W,
                                           ws_mx, ws_mn, ws_psum, ws_pssq);
    pfn_stats<<<1, 64, 0, stream>>>(ws_psum, ws_pssq, gamma, beta, ws_scl, ws_shf);
    pfn_finish<<<(P_PILLARS * COUT) / 256, 256, 0, stream>>>(ws_mx, ws_mn,
                                                             ws_scl, ws_shf, out);
}